// LaplacianLoss_79912161509935
// MI455X (gfx1250) — compile-verified
//
#include <hip/hip_runtime.h>

// ---------------------------------------------------------------------------
// LaplacianLoss on MI455X (gfx1250): HBM-bound stream of the 1 GiB dense L,
// math done on the matrix pipe via V_WMMA_F32_16X16X4_F32 (full f32).
//   y[i, 0:12] = sum_j L[i,j] * x[b,j,d]   (col c = b*3+d)
//   y[i, 12]   = sum_j L[i,j]              (degree, via a ones column in B)
// loss = mean over (b,i,d) of (y[i,c]/(deg_i+1e-6) - x[b,i,d])^2
// ---------------------------------------------------------------------------

#define NV 16384
#define BATCH 4
#define KSEG 4                    // split K dimension across 4 waves per strip
#define KSEG_LEN (NV / KSEG)      // 4096
#define NSTRIP (NV / 16)          // 1024 row-strips of 16 rows
#define WAVES_PER_BLK 8           // 256 threads = 8 wave32
#define WMMA_BLOCKS (NSTRIP * KSEG / WAVES_PER_BLK)   // 512 blocks
#define LOSS_BLOCKS (NV / 256)    // 64
#define INV_COUNT (1.0f / (float)(BATCH * NV * 3))

typedef float v2f __attribute__((ext_vector_type(2)));
typedef float v8f __attribute__((ext_vector_type(8)));

// ---------------------------------------------------------------------------
// Kernel 1: build packed transposed B operand  XpT[16][NV]:
//   rows 0..11 : XpT[c][j] = x[b, j, d]  with c = b*3 + d
//   row  12    : 1.0  (so WMMA column 12 accumulates the degree)
//   rows 13..15: 0.0
// ---------------------------------------------------------------------------
__global__ void lap_pack_xT(const float* __restrict__ x, float* __restrict__ xpT) {
    int j = blockIdx.x * blockDim.x + threadIdx.x;
#pragma unroll
    for (int n = 0; n < 16; ++n) {
        float v;
        if (n < 12) {
            int b = n / 3, d = n % 3;
            v = x[((size_t)b * NV + j) * 3 + d];
        } else if (n == 12) {
            v = 1.0f;
        } else {
            v = 0.0f;
        }
        xpT[(size_t)n * NV + j] = v;
    }
}

// ---------------------------------------------------------------------------
// Kernel 2: WMMA streaming GEMM.  Each wave owns (strip, kseg):
//   16 rows of L  x  4096 K values  ->  16x16 f32 partial into ws.
// A-tile layout (ISA 7.12.2, 32-bit A 16x4): lane<16 holds K0,K1 of row
// M=lane; lane>=16 holds K2,K3 of row M=lane-16  => one b64 load per lane.
// B-tile: lane<16 holds K0,K1 of column N=lane; lane>=16 holds K2,K3
//   => one b64 load per lane from XpT (transposed => contiguous pair).
// ---------------------------------------------------------------------------
__global__ void __launch_bounds__(256)
lap_wmma(const float* __restrict__ Lap, const float* __restrict__ xpT,
         float* __restrict__ partial) {
    const int tid  = threadIdx.x;
    const int lane = tid & 31;
    const int w    = blockIdx.x * WAVES_PER_BLK + (tid >> 5);
    const int strip = w >> 2;          // 0..1023
    const int kseg  = w & (KSEG - 1);  // 0..3
    const int rowbase = strip * 16;
    const int kstart  = kseg * KSEG_LEN;

    const int lmod = lane & 15;
    const int lhi2 = (lane >> 4) * 2;  // 0 for lanes 0-15, 2 for lanes 16-31

    const float* Arow = Lap + (size_t)(rowbase + lmod) * NV + lhi2;
    const float* Brow = xpT + (size_t)lmod * NV + lhi2;

    v8f acc0 = (v8f)0.0f;
    v8f acc1 = (v8f)0.0f;

    for (int kb = kstart; kb < kstart + KSEG_LEN; kb += 64) {
        // prefetch the HBM (A) stream ~2 iterations ahead
        __builtin_prefetch((const void*)(Arow + kb + 128), 0, 1);

        v2f a[16], b[16];
#pragma unroll
        for (int t = 0; t < 16; ++t) {
            a[t] = *(const v2f*)(Arow + kb + 4 * t);   // b64, HBM stream
            b[t] = *(const v2f*)(Brow + kb + 4 * t);   // b64, L2-resident
        }
        // two accumulator chains to break WMMA->WMMA RAW serialization
#pragma unroll
        for (int t = 0; t < 16; t += 2) {
            acc0 = __builtin_amdgcn_wmma_f32_16x16x4_f32(
                false, a[t],     false, b[t],     (short)0, acc0, false, false);
            acc1 = __builtin_amdgcn_wmma_f32_16x16x4_f32(
                false, a[t + 1], false, b[t + 1], (short)0, acc1, false, false);
        }
    }

    // C/D layout: VGPR r, lane l -> element (M = r + (l>>4)*8, N = l&15)
    float* pbase = partial +
        ((size_t)kseg * NV + rowbase + (lane >> 4) * 8) * 16 + lmod;
#pragma unroll
    for (int r = 0; r < 8; ++r)
        pbase[(size_t)r * 16] = acc0[r] + acc1[r];
}

// ---------------------------------------------------------------------------
// Kernel 3: per-row loss terms + per-block reduction (deterministic).
// ---------------------------------------------------------------------------
__global__ void __launch_bounds__(256)
lap_loss(const float* __restrict__ partial, const float* __restrict__ x,
         float* __restrict__ blockPart) {
    const int i = blockIdx.x * blockDim.x + threadIdx.x;   // row / vertex

    float y[16];
#pragma unroll
    for (int c = 0; c < 16; ++c) y[c] = 0.0f;
#pragma unroll
    for (int s = 0; s < KSEG; ++s) {
        const float* p = partial + ((size_t)s * NV + i) * 16;
#pragma unroll
        for (int c = 0; c < 16; ++c) y[c] += p[c];
    }

    const float inv = 1.0f / (y[12] + 1e-6f);
    float acc = 0.0f;
#pragma unroll
    for (int c = 0; c < 12; ++c) {
        int b = c / 3, d = c % 3;
        float s = y[c] * inv - x[((size_t)b * NV + i) * 3 + d];
        acc += s * s;
    }

    __shared__ float red[256];
    red[threadIdx.x] = acc;
    __syncthreads();
#pragma unroll
    for (int off = 128; off > 0; off >>= 1) {
        if (threadIdx.x < off) red[threadIdx.x] += red[threadIdx.x + off];
        __syncthreads();
    }
    if (threadIdx.x == 0) blockPart[blockIdx.x] = red[0];
}

// ---------------------------------------------------------------------------
// Kernel 4: final reduction of 64 block partials -> scalar mean.
// ---------------------------------------------------------------------------
__global__ void lap_final(const float* __restrict__ blockPart, float* __restrict__ out) {
    __shared__ float red[64];
    red[threadIdx.x] = blockPart[threadIdx.x];
    __syncthreads();
#pragma unroll
    for (int off = 32; off > 0; off >>= 1) {
        if (threadIdx.x < off) red[threadIdx.x] += red[threadIdx.x + off];
        __syncthreads();
    }
    if (threadIdx.x == 0) out[0] = red[0] * INV_COUNT;
}

// ---------------------------------------------------------------------------
extern "C" void kernel_launch(void* const* d_in, const int* in_sizes, int n_in,
                              void* d_out, int out_size, void* d_ws, size_t ws_size,
                              hipStream_t stream) {
    const float* x   = (const float*)d_in[0];   // [B, NV, 3] f32
    const float* Lap = (const float*)d_in[1];   // [NV, NV]  f32
    float* out = (float*)d_out;

    // workspace layout
    char* ws = (char*)d_ws;
    float* xpT       = (float*)ws;                                   // 16*NV f32   (1 MB)
    float* partial   = (float*)(ws + (size_t)16 * NV * sizeof(float));      // KSEG*NV*16 (4 MB)
    float* blockPart = (float*)(ws + (size_t)16 * NV * sizeof(float)
                                   + (size_t)KSEG * NV * 16 * sizeof(float)); // 64 f32

    lap_pack_xT<<<NV / 256, 256, 0, stream>>>(x, xpT);
    lap_wmma<<<WMMA_BLOCKS, 256, 0, stream>>>(Lap, xpT, partial);
    lap_loss<<<LOSS_BLOCKS, 256, 0, stream>>>(partial, x, blockPart);
    lap_final<<<1, 64, 0, stream>>>(blockPart, out);
}